// BEVDet4D_29626684408007
// MI455X (gfx1250) — compile-verified
//
#include <hip/hip_runtime.h>
#include <hip/hip_bf16.h>
#include <math.h>

// ---------------------------------------------------------------------------
// BEVDet4D forward for MI455X (gfx1250), wave32, fp32 WMMA (16x16x4).
// Stage-1 GEMM stages its A tile through LDS with gfx1250 async-to-LDS copies
// (ASYNCcnt), then feeds the matrix pipe from ds_load_b64.
// ---------------------------------------------------------------------------

typedef __attribute__((ext_vector_type(2))) float v2f;
typedef __attribute__((ext_vector_type(8))) float v8f;

#define NIMG    24          // B*N
#define HWIMG   704         // 16*44
#define NPOS    16896       // NIMG*HWIMG
#define KPATCH  768         // 3*16*16
#define CMID    256
#define NDCH    105         // D + C_CTX
#define NDPITCH 112         // padded to 7*16
#define DBINS   41
#define CCTX    64
#define NPTS    692736      // NIMG*DBINS*HWIMG
#define BEVN    16384       // 128*128
#define OUTCH   7
#define OUTELEMS (OUTCH*BEVN)   // 114688
#define APITCH  772         // 768 + 4: lanes hit distinct LDS banks, 16B aligned

// workspace layout (floats)
#define OFF_X1     ((size_t)0)                       // NPOS*CMID
#define OFF_Y      (OFF_X1 + (size_t)NPOS*CMID)      // NPOS*NDPITCH
#define OFF_DEPTH  (OFF_Y  + (size_t)NPOS*NDPITCH)   // NPTS  (layout (bn,d,hw))
#define OFF_CTX    (OFF_DEPTH + (size_t)NPTS)        // CCTX*NPOS (layout (c,p))
#define OFF_FEAT   (OFF_CTX + (size_t)CCTX*NPOS)     // BEVN*CCTX (layout (p,c))
#define OFF_WFOLD  (OFF_FEAT + (size_t)BEVN*CCTX)    // 64*576
#define OFF_TF     (OFF_WFOLD + (size_t)64*576)      // 24*18

__device__ __forceinline__ v8f wmma_f32_k4(v2f a, v2f b, v8f c) {
    return __builtin_amdgcn_wmma_f32_16x16x4_f32(
        /*neg_a=*/false, a, /*neg_b=*/false, b,
        /*c_mod=*/(short)0, c, /*reuse_a=*/false, /*reuse_b=*/false);
}

__device__ __forceinline__ void inv3(const float* m, float* o) {
    float a=m[0],b=m[1],c=m[2],d=m[3],e=m[4],f=m[5],g=m[6],h=m[7],i=m[8];
    float A =  (e*i - f*h), B = -(d*i - f*g), C =  (d*h - e*g);
    float det = a*A + b*B + c*C;
    float inv = 1.0f / det;
    o[0] = A*inv;              o[1] = -(b*i - c*h)*inv;  o[2] =  (b*f - c*e)*inv;
    o[3] = B*inv;              o[4] =  (a*i - c*g)*inv;  o[5] = -(a*f - c*d)*inv;
    o[6] = C*inv;              o[7] = -(a*h - b*g)*inv;  o[8] =  (a*e - b*d)*inv;
}

// -------- per-camera transforms: inv(post_rots), rots @ inv(intrins) --------
__global__ void bev_tf_kernel(const float* __restrict__ rots,
                              const float* __restrict__ intrins,
                              const float* __restrict__ post_rots,
                              float* __restrict__ tf) {
    int i = threadIdx.x;
    if (i >= NIMG) return;
    float pr[9], Km[9], R[9], ipr[9], iK[9];
    for (int j = 0; j < 9; ++j) { pr[j]=post_rots[i*9+j]; Km[j]=intrins[i*9+j]; R[j]=rots[i*9+j]; }
    inv3(pr, ipr);
    inv3(Km, iK);
    float* o = tf + i*18;
    for (int j = 0; j < 9; ++j) o[j] = ipr[j];
    for (int r = 0; r < 3; ++r)
        for (int c = 0; c < 3; ++c)
            o[9 + r*3 + c] = R[r*3+0]*iK[0*3+c] + R[r*3+1]*iK[1*3+c] + R[r*3+2]*iK[2*3+c];
}

// -------- fold concat([bev,bev]) into the 3x3 conv weights --------
__global__ void bev_wfold_kernel(const float* __restrict__ enc_w, float* __restrict__ wf) {
    int t = blockIdx.x * blockDim.x + threadIdx.x;
    if (t >= 64*576) return;
    int o = t / 576, k = t % 576;
    int c = k / 9,  tap = k % 9;
    wf[t] = enc_w[((size_t)(o*128 + c))*9 + tap] + enc_w[((size_t)(o*128 + c + 64))*9 + tap];
}

__global__ void bev_zero_kernel(float* __restrict__ p, int n) {
    int t = blockIdx.x * blockDim.x + threadIdx.x;
    if (t < n) p[t] = 0.0f;
}

// -------- stage 1: 16x16/s16 patch conv as WMMA GEMM, fused BN1 + ReLU -------
// One 256-thread block (8 waves) per M-tile of 16 positions. The 16x768 fp32
// A tile is staged into LDS once via async-to-LDS B128 copies; each wave then
// computes two 16-channel N tiles (8 waves x 2 = 256 channels) from LDS.
__global__ __launch_bounds__(256)
void bev_conv1_wmma(const float* __restrict__ imgs, const float* __restrict__ convw,
                    const float* __restrict__ convb,
                    const float* __restrict__ g1, const float* __restrict__ b1,
                    const float* __restrict__ m1, const float* __restrict__ v1,
                    float* __restrict__ x1) {
    __shared__ float tileA[16 * APITCH];
    const int t  = threadIdx.x;
    const int p0 = blockIdx.x * 16;

    // ---- async staging: 16 rows x 768 K-values = 3072 B128 transfers -------
    // K index decomposes as k = c*256 + py*16 + px; each (m,c,py) row is 16
    // contiguous floats in the image -> four 16-byte chunks.
#pragma unroll
    for (int i = 0; i < 12; ++i) {
        int tau = t + (i << 8);           // 0..3071
        int row = tau >> 2;               // (m, c, py)
        int q   = tau & 3;                // 16B chunk in row
        int m   = row / 48;
        int r   = row % 48;
        int c   = r >> 4;
        int py  = r & 15;
        int px  = q << 2;
        int p   = p0 + m;
        int bn = p / HWIMG, pos = p % HWIMG;
        int oh = pos / 44,  ow  = pos % 44;
        const float* g = imgs + ((size_t)(bn*3 + c)*256 + oh*16 + py)*704 + ow*16 + px;
        unsigned lo = (unsigned)(size_t)(const void*)&tileA[m*APITCH + c*256 + py*16 + px];
        asm volatile("global_load_async_to_lds_b128 %0, %1, off"
                     :: "v"(lo), "v"(g) : "memory");
    }
    asm volatile("s_wait_asynccnt 0" ::: "memory");
    __syncthreads();

    // ---- matrix pipe: fp32 WMMA 16x16x4, A from LDS, B from L2 -------------
    const int wave = t >> 5;
    const int lane = t & 31;
    const int hi   = lane >> 4;           // 0: K0/K1  1: K2/K3
    const int col  = lane & 15;           // A row == B col == C/D col
    const float* aLds = &tileA[col * APITCH];

    for (int rep = 0; rep < 2; ++rep) {
        int n0 = (wave * 2 + rep) << 4;
        int ch = n0 + col;
        const float* wrow = convw + (size_t)ch * KPATCH;

        v8f acc = {};
        for (int k = 0; k < KPATCH; k += 4) {
            int kk = k + (hi << 1);
            v2f a; a.x = aLds[kk];   a.y = aLds[kk + 1];   // ds_load_b64
            v2f b; b.x = wrow[kk];   b.y = wrow[kk + 1];
            acc = wmma_f32_k4(a, b, acc);
        }

        float sc = g1[ch] / sqrtf(v1[ch] + 1e-5f);
        float sh = b1[ch] - m1[ch] * sc;
        float cb = convb[ch];
#pragma unroll
        for (int r = 0; r < 8; ++r) {
            int row = r + (hi << 3);
            float v = (acc[r] + cb) * sc + sh;
            x1[(size_t)(p0 + row) * CMID + ch] = fmaxf(v, 0.0f);
        }
    }
}

// -------- stage 2: depth/ctx head GEMM: M=16896, K=256, N=105 (pad 112) -----
__global__ __launch_bounds__(32)
void bev_depth_wmma(const float* __restrict__ x1, const float* __restrict__ dw,
                    const float* __restrict__ db, float* __restrict__ y) {
    int lane = threadIdx.x;
    int hi   = lane >> 4;
    int col  = lane & 15;
    int p0   = blockIdx.x * 16;
    int n0   = blockIdx.y * 16;

    const float* arow = x1 + (size_t)(p0 + col) * CMID;
    int ch = n0 + col;
    const float* brow = dw + (size_t)(ch < NDCH ? ch : 0) * CMID;
    float bmask = (ch < NDCH) ? 1.0f : 0.0f;

    v8f acc = {};
    for (int k = 0; k < CMID; k += 4) {
        int kk = k + (hi << 1);
        v2f a; a.x = arow[kk];          a.y = arow[kk + 1];
        v2f b; b.x = brow[kk] * bmask;  b.y = brow[kk + 1] * bmask;
        acc = wmma_f32_k4(a, b, acc);
    }

    if (ch < NDCH) {
        float bias = db[ch];
#pragma unroll
        for (int r = 0; r < 8; ++r) {
            int row = r + (hi << 3);
            y[(size_t)(p0 + row) * NDPITCH + ch] = acc[r] + bias;
        }
    }
}

// -------- stage 3: softmax over 41 depth bins, split depth / ctx ------------
__global__ void bev_softmax_kernel(const float* __restrict__ y,
                                   float* __restrict__ depth, float* __restrict__ ctx) {
    int p = blockIdx.x * blockDim.x + threadIdx.x;
    if (p >= NPOS) return;
    const float* yp = y + (size_t)p * NDPITCH;
    float e[DBINS];
    float mx = yp[0];
    for (int d = 1; d < DBINS; ++d) mx = fmaxf(mx, yp[d]);
    float s = 0.0f;
    for (int d = 0; d < DBINS; ++d) { e[d] = __expf(yp[d] - mx); s += e[d]; }
    float inv = 1.0f / s;
    int bn = p / HWIMG, hw = p % HWIMG;
    for (int d = 0; d < DBINS; ++d)
        depth[((size_t)(bn*DBINS + d)) * HWIMG + hw] = e[d] * inv;   // (bn,d,hw)
    for (int c = 0; c < CCTX; ++c)
        ctx[(size_t)c * NPOS + p] = yp[DBINS + c];                   // (c,p)
}

// -------- stage 4: geometry + voxel pooling (scatter-add into bev) ----------
__global__ void bev_pool_kernel(const float* __restrict__ tf,
                                const float* __restrict__ trans,
                                const float* __restrict__ post_trans,
                                const float* __restrict__ depth,
                                const float* __restrict__ ctx,
                                float* __restrict__ bev) {
    int t = blockIdx.x * blockDim.x + threadIdx.x;
    if (t >= NPTS) return;
    int bn  = t / (DBINS * HWIMG);
    int rem = t % (DBINS * HWIMG);
    int d   = rem / HWIMG;
    int hw  = rem % HWIMG;
    int h   = hw / 44, w = hw % 44;

    const float* T = tf + bn * 18;
    float fx = (float)w * (703.0f / 43.0f);
    float fy = (float)h * 17.0f;
    float fz = 4.0f + (float)d;
    float vx = fx - post_trans[bn*3 + 0];
    float vy = fy - post_trans[bn*3 + 1];
    float vz = fz - post_trans[bn*3 + 2];
    float ux = T[0]*vx + T[1]*vy + T[2]*vz;
    float uy = T[3]*vx + T[4]*vy + T[5]*vz;
    float uz = T[6]*vx + T[7]*vy + T[8]*vz;
    ux *= uz; uy *= uz;
    float qx = T[ 9]*ux + T[10]*uy + T[11]*uz + trans[bn*3 + 0];
    float qy = T[12]*ux + T[13]*uy + T[14]*uz + trans[bn*3 + 1];
    float qz = T[15]*ux + T[16]*uy + T[17]*uz + trans[bn*3 + 2];

    int cx = (int)((qx + 51.2f) / 0.8f);
    int cy = (int)((qy + 51.2f) / 0.8f);
    int cz = (int)((qz + 10.0f) / 20.0f);
    int rank = cx + cy * 128 + cz * BEVN;     // reference checks rank only
    if (rank < 0 || rank >= BEVN) return;

    float dv  = depth[t];                     // exact (bn,d,hw) layout
    int pimg  = bn * HWIMG + hw;
    const float* cp = ctx + pimg;
#pragma unroll 8
    for (int c = 0; c < CCTX; ++c)
        unsafeAtomicAdd(&bev[(size_t)c * BEVN + rank], cp[(size_t)c * NPOS] * dv);
}

// -------- stage 5: 3x3 conv as WMMA implicit GEMM, fused BN2 + ReLU ---------
// M = 16384 pixels, K = 64*9 = 576 (concat folded), N = 64.
__global__ __launch_bounds__(32)
void bev_conv2_wmma(const float* __restrict__ bev, const float* __restrict__ wf,
                    const float* __restrict__ encb,
                    const float* __restrict__ g2, const float* __restrict__ b2,
                    const float* __restrict__ m2, const float* __restrict__ v2,
                    float* __restrict__ feat) {
    int lane = threadIdx.x;
    int hi   = lane >> 4;
    int col  = lane & 15;
    int p0   = blockIdx.x * 16;
    int n0   = blockIdx.y * 16;

    int pm = p0 + col;
    int yy0 = pm >> 7, xx0 = pm & 127;
    const float* wrow = wf + (size_t)(n0 + col) * 576;

    v8f acc = {};
    for (int k = 0; k < 576; k += 4) {
        int kk = k + (hi << 1);
        v2f a, b;
#pragma unroll
        for (int j = 0; j < 2; ++j) {
            int kj  = kk + j;
            int c   = kj / 9;
            int tap = kj - c * 9;
            int ky  = tap / 3 - 1;
            int kx  = tap - (tap / 3) * 3 - 1;
            int yy = yy0 + ky, xx = xx0 + kx;
            float av = (yy >= 0 && yy < 128 && xx >= 0 && xx < 128)
                           ? bev[(size_t)c * BEVN + yy * 128 + xx] : 0.0f;
            if (j == 0) a.x = av; else a.y = av;
        }
        b.x = wrow[kk]; b.y = wrow[kk + 1];
        acc = wmma_f32_k4(a, b, acc);
    }

    int ch = n0 + col;
    float sc = g2[ch] / sqrtf(v2[ch] + 1e-5f);
    float sh = b2[ch] - m2[ch] * sc;
    float cb = encb[ch];
#pragma unroll
    for (int r = 0; r < 8; ++r) {
        int row = r + (hi << 3);
        float v = (acc[r] + cb) * sc + sh;
        feat[(size_t)(p0 + row) * CCTX + ch] = fmaxf(v, 0.0f);
    }
}

// -------- stage 6: 7x64 head -------------------------------------------------
__global__ void bev_head_kernel(const float* __restrict__ feat,
                                const float* __restrict__ hw_, const float* __restrict__ hb,
                                float* __restrict__ out) {
    int t = blockIdx.x * blockDim.x + threadIdx.x;
    if (t >= OUTELEMS) return;
    int o = t / BEVN, p = t % BEVN;
    const float* fp = feat + (size_t)p * CCTX;
    const float* wp = hw_ + o * CCTX;
    float s = hb[o];
#pragma unroll 8
    for (int c = 0; c < CCTX; ++c) s += wp[c] * fp[c];
    out[(size_t)o * BEVN + p] = s;
}

extern "C" void kernel_launch(void* const* d_in, const int* in_sizes, int n_in,
                              void* d_out, int out_size, void* d_ws, size_t ws_size,
                              hipStream_t stream) {
    const float* imgs      = (const float*)d_in[0];
    const float* rots      = (const float*)d_in[1];
    const float* trans     = (const float*)d_in[2];
    const float* intrins   = (const float*)d_in[3];
    const float* post_rots = (const float*)d_in[4];
    const float* post_trans= (const float*)d_in[5];
    const float* conv_w    = (const float*)d_in[6];
    const float* conv_b    = (const float*)d_in[7];
    const float* bn1_g     = (const float*)d_in[8];
    const float* bn1_b     = (const float*)d_in[9];
    const float* bn1_m     = (const float*)d_in[10];
    const float* bn1_v     = (const float*)d_in[11];
    const float* depth_w   = (const float*)d_in[12];
    const float* depth_b   = (const float*)d_in[13];
    const float* enc_w     = (const float*)d_in[14];
    const float* enc_b     = (const float*)d_in[15];
    const float* bn2_g     = (const float*)d_in[16];
    const float* bn2_b     = (const float*)d_in[17];
    const float* bn2_m     = (const float*)d_in[18];
    const float* bn2_v     = (const float*)d_in[19];
    const float* head_w    = (const float*)d_in[20];
    const float* head_b    = (const float*)d_in[21];

    float* ws    = (float*)d_ws;
    float* x1    = ws + OFF_X1;
    float* yws   = ws + OFF_Y;
    float* dep   = ws + OFF_DEPTH;
    float* ctx   = ws + OFF_CTX;
    float* feat  = ws + OFF_FEAT;
    float* wfold = ws + OFF_WFOLD;
    float* tf    = ws + OFF_TF;

    float* out = (float*)d_out;           // (1,7,128,128)
    float* bev = out + OUTELEMS;          // (1,64,128,128)

    bev_tf_kernel<<<1, 32, 0, stream>>>(rots, intrins, post_rots, tf);
    bev_wfold_kernel<<<(64*576 + 255)/256, 256, 0, stream>>>(enc_w, wfold);
    bev_zero_kernel<<<(CCTX*BEVN + 255)/256, 256, 0, stream>>>(bev, CCTX*BEVN);

    bev_conv1_wmma<<<NPOS/16, 256, 0, stream>>>(
        imgs, conv_w, conv_b, bn1_g, bn1_b, bn1_m, bn1_v, x1);

    bev_depth_wmma<<<dim3(NPOS/16, NDPITCH/16), 32, 0, stream>>>(x1, depth_w, depth_b, yws);

    bev_softmax_kernel<<<(NPOS + 255)/256, 256, 0, stream>>>(yws, dep, ctx);

    bev_pool_kernel<<<(NPTS + 255)/256, 256, 0, stream>>>(tf, trans, post_trans, dep, ctx, bev);

    bev_conv2_wmma<<<dim3(BEVN/16, CCTX/16), 32, 0, stream>>>(
        bev, wfold, enc_b, bn2_g, bn2_b, bn2_m, bn2_v, feat);

    bev_head_kernel<<<(OUTELEMS + 255)/256, 256, 0, stream>>>(feat, head_w, head_b, out);
}